// FaVoRmodel_51934744543702
// MI455X (gfx1250) — compile-verified
//
#include <hip/hip_runtime.h>

// ---------------------------------------------------------------------------
// Volume-rendering weights (Raw2Alpha + Alphas2Weights + composite), MI455X.
// One wave32 per ray (ray_id sorted -> contiguous segments, ~64 samples/ray).
// Memory-bound (~190MB @ 23.3TB/s ~= 9us); WMMA inapplicable (no matmul).
// CDNA5 features: global_load_async_to_lds_b32 double buffering (ASYNCcnt +
// s_wait_asynccnt), wave32 ballot 32-ary segment search, shuffle scan.
// ---------------------------------------------------------------------------

#define WAVE   32
#define WPB    8                 // waves per block
#define BLOCK  (WAVE * WPB)
#define CHUNK  32                // samples processed per wave per iteration
#define SLICE  128               // floats per LDS buffer: 32 density + 96 rgb

#define INTERVAL 0.5f

// AS3 (LDS) byte address == low 32 bits of the generic shared-aperture address.
__device__ __forceinline__ unsigned lds_off(const void* p) {
  return (unsigned)(unsigned long long)p;
}

// CDNA5 async copy: each lane copies 4 bytes from the global address in vaddr to the LDS address in vdst, tracked by ASYNCcnt.
// GV mode: vdst = per-lane LDS byte address, vaddr = per-lane 64-bit address.
__device__ __forceinline__ void async_gload_b32(const float* g, unsigned l) {
  asm volatile("global_load_async_to_lds_b32 %0, %1, off"
               :
               : "v"(l), "v"((unsigned long long)(const void*)g)
               : "memory");
}

__device__ __forceinline__ void wait_async_le4() {
  asm volatile("s_wait_asynccnt 4" ::: "memory");
}
__device__ __forceinline__ void wait_async_le0() {
  asm volatile("s_wait_asynccnt 0" ::: "memory");
}

// Wave-cooperative lower_bound on a sorted int array: 32 lanes probe 32
// equally spaced split points per step (range shrinks 32x per step, ~5 steps
// for 8.4M instead of ~23 serial binary-search round trips). All probes hit
// the 32MB ray_id array, which is resident in the 192MB L2.
__device__ __forceinline__ int wave_lower_bound(const int* __restrict__ a,
                                                int lo, int hi, int t,
                                                int lane) {
  while (hi - lo > WAVE) {
    const int span = hi - lo;
    const int cw   = (span + WAVE - 1) >> 5;    // chunk width >= 1
    const int p    = lo + lane * cw;
    const bool c   = (p < hi) && (a[p] < t);    // monotone non-increasing in lane
    const int  j   = __popcll(__ballot(c));     // lanes 0..j-1 are true
    if (j == 0) return lo;                      // a[lo] >= t -> answer == lo
    int nlo = lo + (j - 1) * cw + 1;            // a[p_{j-1}] < t
    int nhi = lo + j * cw;                      // first probe with a[p] >= t
    if (nhi > hi) nhi = hi;
    lo = nlo;
    hi = nhi;
  }
  const int  p = lo + lane;
  const bool c = (p < hi) && (a[p] < t);
  return lo + __popcll(__ballot(c));
}

// Stage one chunk (32 samples: 32 density + 96 rgb floats) into an LDS buffer.
// Indices are clamped (never predicated) so EXEC stays all-ones and padding
// lanes fetch harmless in-range data; their alpha is forced to 0 later.
__device__ __forceinline__ void prefetch_chunk(const float* __restrict__ density,
                                               const float* __restrict__ rgb,
                                               int base, int lane, int M,
                                               float* buf) {
  int i = base + lane;           if (i > M - 1) i = M - 1;
  int rb = base * 3;
  int mx = M * 3 - 1;
  int j0 = rb + lane;            if (j0 > mx) j0 = mx;
  int j1 = rb + lane + 32;       if (j1 > mx) j1 = mx;
  int j2 = rb + lane + 64;       if (j2 > mx) j2 = mx;
  async_gload_b32(&density[i], lds_off(&buf[lane]));
  async_gload_b32(&rgb[j0],    lds_off(&buf[32 + lane]));
  async_gload_b32(&rgb[j1],    lds_off(&buf[64 + lane]));
  async_gload_b32(&rgb[j2],    lds_off(&buf[96 + lane]));
}

__global__ void __launch_bounds__(BLOCK)
favor_march_kernel(const float* __restrict__ density,
                   const float* __restrict__ rgb,
                   const float* __restrict__ shift,
                   const int*   __restrict__ ray_id,
                   float* __restrict__ outW,      // [M] weights
                   float* __restrict__ outAinv,   // [N] alphainv_last
                   float* __restrict__ outRGB,    // [N,3] composite
                   int M, int N) {
  __shared__ float lds[WPB * 2 * SLICE];          // 8KB/block, dbl-buffered

  const int lane = threadIdx.x & (WAVE - 1);
  const int wv   = threadIdx.x >> 5;
  const int r    = blockIdx.x * WPB + wv;         // one ray per wave
  if (r >= N) return;

  // Segment bounds via wave-cooperative 32-ary search; second search seeded
  // with s0 (segments are sorted).
  const int s0 = wave_lower_bound(ray_id, 0,  M, r,     lane);
  const int s1 = wave_lower_bound(ray_id, s0, M, r + 1, lane);

  float* slice = &lds[wv * 2 * SLICE];
  const float sh = shift[0];
  const int nChunks = (s1 - s0 + CHUNK - 1) / CHUNK;

  if (nChunks > 0) prefetch_chunk(density, rgb, s0, lane, M, slice);

  float c  = 0.f;                 // running log-transmittance carry (uniform)
  float ar = 0.f, ag = 0.f, ab = 0.f;

  for (int k = 0; k < nChunks; ++k) {
    const int base = s0 + k * CHUNK;
    float* buf = slice + (k & 1) * SLICE;

    if (k + 1 < nChunks) {
      prefetch_chunk(density, rgb, base + CHUNK, lane, M,
                     slice + ((k + 1) & 1) * SLICE);
      wait_async_le4();           // in-order completion: chunk k is resident
    } else {
      wait_async_le0();
    }

    const int  i   = base + lane;
    const bool act = (i < s1);

    // Raw2Alpha in log space: log1ma = -softplus(d+shift) * interval
    float x  = buf[lane] + sh;
    float sp = fmaxf(x, 0.f) + log1pf(__expf(-fabsf(x)));
    float l  = act ? (-INTERVAL * sp) : 0.f;      // padding lanes: identity

    // Wave32 inclusive scan of l (5 shuffle steps)
    float incl = l;
    #pragma unroll
    for (int d = 1; d < WAVE; d <<= 1) {
      float t = __shfl_up(incl, d, WAVE);
      if (lane >= d) incl += t;
    }

    float T     = __expf(c + (incl - l));         // exclusive + carry
    float alpha = 1.f - __expf(l);                // 0 exactly for padding
    float w     = alpha * T;
    if (act) outW[i] = w;

    // composite: sigmoid(rgb) weighted (w==0 kills padding-lane garbage)
    float sr = 1.f / (1.f + __expf(-buf[32 + lane * 3 + 0]));
    float sg = 1.f / (1.f + __expf(-buf[32 + lane * 3 + 1]));
    float sb = 1.f / (1.f + __expf(-buf[32 + lane * 3 + 2]));
    ar += w * sr;  ag += w * sg;  ab += w * sb;

    c += __shfl(incl, WAVE - 1, WAVE);            // fold chunk total into carry
  }

  // Butterfly reduction of rgb accumulators across the wave
  #pragma unroll
  for (int d = WAVE / 2; d > 0; d >>= 1) {
    ar += __shfl_xor(ar, d, WAVE);
    ag += __shfl_xor(ag, d, WAVE);
    ab += __shfl_xor(ab, d, WAVE);
  }

  if (lane == 0) {
    float ainv = __expf(c);                       // empty ray -> exp(0) = 1
    outAinv[r]        = ainv;
    outRGB[3 * r + 0] = ar + ainv;
    outRGB[3 * r + 1] = ag + ainv;
    outRGB[3 * r + 2] = ab + ainv;
  }
}

extern "C" void kernel_launch(void* const* d_in, const int* in_sizes, int n_in,
                              void* d_out, int out_size, void* d_ws, size_t ws_size,
                              hipStream_t stream) {
  (void)n_in; (void)d_ws; (void)ws_size;
  const float* density = (const float*)d_in[0];
  const float* rgb     = (const float*)d_in[1];
  const float* shift   = (const float*)d_in[2];
  const int*   ray_id  = (const int*)d_in[3];

  const int M = in_sizes[0];                 // samples
  const int N = (out_size - M) / 4;          // out = [M] + [N] + [N,3]

  float* outW    = (float*)d_out;
  float* outAinv = outW + M;
  float* outRGB  = outAinv + N;

  const int blocks = (N + WPB - 1) / WPB;    // one wave32 per ray
  favor_march_kernel<<<blocks, BLOCK, 0, stream>>>(
      density, rgb, shift, ray_id, outW, outAinv, outRGB, M, N);
}